// DecoderLSTM_30580167147635
// MI455X (gfx1250) — compile-verified
//
#include <hip/hip_runtime.h>
#include <math.h>

typedef __attribute__((ext_vector_type(2))) float v2f;
typedef __attribute__((ext_vector_type(4))) float v4f;
typedef __attribute__((ext_vector_type(8))) float v8f;

#define BATCH 128
#define NENC  8192
#define HD    256
#define HD3   768
#define HD4   1024
#define STEPS 10
#define SPLIT 32   // workgroups per batch in attention kernel

// workspace layout (floats)
#define OFF_X 0u                          // BATCH*HD3 = 98304
#define OFF_H 98304u                      // BATCH*HD  = 32768
#define OFF_C 131072u                     // BATCH*HD  = 32768
#define OFF_G 163840u                     // BATCH*HD4 = 131072
#define OFF_Q 294912u                     // BATCH*HD  = 32768

__device__ __forceinline__ float sigf(float x) { return 1.0f / (1.0f + expf(-x)); }

__device__ __forceinline__ v8f wmma_f32(v2f a, v2f b, v8f c) {
  // V_WMMA_F32_16X16X4_F32 : D = A(16x4 f32) * B(4x16 f32) + C(16x16 f32)
  return __builtin_amdgcn_wmma_f32_16x16x4_f32(false, a, false, b, (short)0, c, false, false);
}

// ---------------------------------------------------------------- init state
__global__ __launch_bounds__(256)
void k_init(const float* __restrict__ hidden, const float* __restrict__ cell,
            const float* __restrict__ x0, float* __restrict__ ws) {
  int i = blockIdx.x * 256 + threadIdx.x;
  if (i < BATCH * HD)  { ws[OFF_H + i] = hidden[i]; ws[OFF_C + i] = cell[i]; }
  if (i < BATCH * HD3) { ws[OFF_X + i] = x0[i]; }
}

// ------------------------------------------------------- LSTM gates (WMMA f32)
// gates(128x1024) = x(128x768)@W_ih^T + h(128x256)@W_hh^T + b_ih + b_hh
// 512 wave-tiles: one 16x16 tile per wave, K-loop in steps of 4.
__global__ __launch_bounds__(256)
void k_gates(const float* __restrict__ x, const float* __restrict__ h,
             const float* __restrict__ Wih, const float* __restrict__ Whh,
             const float* __restrict__ bih, const float* __restrict__ bhh,
             float* __restrict__ gates) {
  int lane = threadIdx.x & 31;
  int wave = blockIdx.x * 8 + (threadIdx.x >> 5);   // 0..511
  int b0 = (wave >> 6) * 16;                        // batch tile
  int j0 = (wave & 63) * 16;                        // gate-column tile
  int row = lane & 15;
  int kk  = (lane >> 4) * 2;                        // 0 or 2

  v8f acc = {};
  const float* ap = x   + (size_t)(b0 + row) * HD3 + kk;
  const float* bp = Wih + (size_t)(j0 + row) * HD3 + kk;
  for (int k = 0; k < HD3; k += 4) {
    v2f a = *(const v2f*)ap; ap += 4;
    v2f b = *(const v2f*)bp; bp += 4;
    acc = wmma_f32(a, b, acc);
  }
  ap = h   + (size_t)(b0 + row) * HD + kk;
  bp = Whh + (size_t)(j0 + row) * HD + kk;
  for (int k = 0; k < HD; k += 4) {
    v2f a = *(const v2f*)ap; ap += 4;
    v2f b = *(const v2f*)bp; bp += 4;
    acc = wmma_f32(a, b, acc);
  }
  // D layout: VGPR v holds M = v + (lane<16 ? 0 : 8), N = lane&15
  int n = lane & 15;
  int mb = (lane >> 4) * 8;
  float bias = bih[j0 + n] + bhh[j0 + n];
  float* gp = gates + (size_t)(b0 + mb) * HD4 + (j0 + n);
#pragma unroll
  for (int v = 0; v < 8; v++) gp[(size_t)v * HD4] = acc[v] + bias;
}

// -------------------------------------------------------- elementwise update
__global__ __launch_bounds__(256)
void k_lstm(const float* __restrict__ gates, float* __restrict__ h, float* __restrict__ c) {
  int idx = blockIdx.x * 256 + threadIdx.x;         // 0..32767
  int b = idx >> 8, d = idx & 255;
  const float* g = gates + (size_t)b * HD4;
  float ig = sigf(g[d]);
  float fg = sigf(g[HD + d]);
  float gg = tanhf(g[2 * HD + d]);
  float og = sigf(g[3 * HD + d]);
  float cn = fg * c[idx] + ig * gg;
  float hn = og * tanhf(cn);
  c[idx] = cn;
  h[idx] = hn;
}

// ------------------------------------------------------- query GEMM (WMMA f32)
// q(128x256) = h(128x256) @ Wq^T(256x256) + bq ; 128 wave-tiles
__global__ __launch_bounds__(256)
void k_query(const float* __restrict__ h, const float* __restrict__ Wq,
             const float* __restrict__ bq, float* __restrict__ q) {
  int lane = threadIdx.x & 31;
  int wave = blockIdx.x * 8 + (threadIdx.x >> 5);   // 0..127
  int b0 = (wave >> 4) * 16;
  int n0 = (wave & 15) * 16;
  int row = lane & 15;
  int kk  = (lane >> 4) * 2;

  v8f acc = {};
  const float* ap = h  + (size_t)(b0 + row) * HD + kk;
  const float* bp = Wq + (size_t)(n0 + row) * HD + kk;
  for (int k = 0; k < HD; k += 4) {
    v2f a = *(const v2f*)ap; ap += 4;
    v2f b = *(const v2f*)bp; bp += 4;
    acc = wmma_f32(a, b, acc);
  }
  int n = lane & 15;
  int mb = (lane >> 4) * 8;
  float bias = bq[n0 + n];
  float* qp = q + (size_t)(b0 + mb) * HD + (n0 + n);
#pragma unroll
  for (int v = 0; v < 8; v++) qp[(size_t)v * HD] = acc[v] + bias;
}

// ------------------------------------------- attention GEMV (the HBM streamer)
// logits[b, n0..n0+15] via WMMA: A = 16-row enc tile, B = q broadcast to 16 cols.
__global__ __launch_bounds__(256)
void k_attn(const float* __restrict__ enc, const float* __restrict__ q,
            float* __restrict__ logits) {
  int b = blockIdx.x / SPLIT;
  int s = blockIdx.x % SPLIT;
  int wave = threadIdx.x >> 5;
  int lane = threadIdx.x & 31;
  int row = lane & 15;
  int kk  = (lane >> 4) * 2;
  const float* qb = q + (size_t)b * HD + kk;

#pragma unroll
  for (int t = 0; t < 2; t++) {
    int n0 = ((s * 8 + wave) * 2 + t) * 16;         // tile 0..511 -> n offset
    const float* ap = enc + ((size_t)b * NENC + n0 + row) * HD + kk;
    const float* bp = qb;
    v8f acc = {};
#pragma unroll 8
    for (int k = 0; k < HD; k += 4) {
      v2f a = __builtin_nontemporal_load((const v2f*)ap); ap += 4;  // 1 GB stream: NT
      v2f bb = *(const v2f*)bp; bp += 4;
      acc = wmma_f32(a, bb, acc);
    }
    // All 16 D columns identical; lane 0 holds logits[n0+0..7], lane 16 holds +8..15
    if (row == 0) {
      float* out = logits + (size_t)b * NENC + n0 + (lane >> 4) * 8;
      v4f lo = {acc[0], acc[1], acc[2], acc[3]};
      v4f hi = {acc[4], acc[5], acc[6], acc[7]};
      *(v4f*)out = lo;
      *(v4f*)(out + 4) = hi;
    }
  }
}

// --------------------------------------------------- top-3 + gather next input
__device__ __forceinline__ bool better(float v, int i, float w, int k) {
  return (v > w) || (v == w && i < k);
}

__global__ __launch_bounds__(256)
void k_topk(const float* __restrict__ logits, const float* __restrict__ enc,
            int* __restrict__ idx_out, float* __restrict__ xn) {
  __shared__ float sv[256][3];
  __shared__ int   si[256][3];
  __shared__ int   topi[3];
  int b = blockIdx.x, t = threadIdx.x;

  float v0 = -INFINITY, v1 = -INFINITY, v2 = -INFINITY;
  int i0 = 0x7fffffff, i1 = 0x7fffffff, i2 = 0x7fffffff;
  const float* lb = logits + (size_t)b * NENC;
  for (int n = t; n < NENC; n += 256) {
    float v = lb[n];                                  // ascending n: strict > keeps earliest index
    if (v > v2) {
      if (v > v0)      { v2=v1; i2=i1; v1=v0; i1=i0; v0=v; i0=n; }
      else if (v > v1) { v2=v1; i2=i1; v1=v;  i1=n; }
      else             { v2=v;  i2=n; }
    }
  }
  sv[t][0]=v0; sv[t][1]=v1; sv[t][2]=v2;
  si[t][0]=i0; si[t][1]=i1; si[t][2]=i2;
  __syncthreads();

  for (int s = 128; s > 0; s >>= 1) {
    if (t < s) {
      float a0=sv[t][0], a1=sv[t][1], a2=sv[t][2];
      int   j0=si[t][0], j1=si[t][1], j2=si[t][2];
#pragma unroll
      for (int j = 0; j < 3; j++) {
        float v = sv[t + s][j]; int i = si[t + s][j];
        if (better(v, i, a0, j0))      { a2=a1; j2=j1; a1=a0; j1=j0; a0=v; j0=i; }
        else if (better(v, i, a1, j1)) { a2=a1; j2=j1; a1=v;  j1=i; }
        else if (better(v, i, a2, j2)) { a2=v;  j2=i; }
      }
      sv[t][0]=a0; sv[t][1]=a1; sv[t][2]=a2;
      si[t][0]=j0; si[t][1]=j1; si[t][2]=j2;
    }
    __syncthreads();
  }

  if (t < 3) {
    int idx = si[0][t];
    topi[t] = idx;
    idx_out[b * 3 + t] = idx;
  }
  __syncthreads();

  // gather 3 encoder rows (768 floats) -> next x
  for (int e = t; e < HD3; e += 256) {
    int j = e >> 8, d = e & 255;
    xn[(size_t)b * HD3 + e] = enc[((size_t)b * NENC + topi[j]) * HD + d];
  }
}

// ---------------------------------------------------------------------- launch
extern "C" void kernel_launch(void* const* d_in, const int* in_sizes, int n_in,
                              void* d_out, int out_size, void* d_ws, size_t ws_size,
                              hipStream_t stream) {
  const float* enc    = (const float*)d_in[0];
  const float* hidden = (const float*)d_in[1];
  const float* cell   = (const float*)d_in[2];
  // d_in[3] end_node_embed: unused in prediction path
  const float* x0     = (const float*)d_in[4];
  const float* Wih    = (const float*)d_in[5];
  const float* bih    = (const float*)d_in[6];
  const float* Whh    = (const float*)d_in[7];
  const float* bhh    = (const float*)d_in[8];
  const float* Wq     = (const float*)d_in[9];
  const float* bq     = (const float*)d_in[10];

  float* ws = (float*)d_ws;
  float* xw = ws + OFF_X;
  float* hw = ws + OFF_H;
  float* cw = ws + OFF_C;
  float* gw = ws + OFF_G;
  float* qw = ws + OFF_Q;

  float* logits_all = (float*)d_out;
  int*   idx_all    = (int*)d_out + (size_t)STEPS * BATCH * NENC;

  k_init<<<(BATCH * HD3 + 255) / 256, 256, 0, stream>>>(hidden, cell, x0, ws);

  for (int s = 0; s < STEPS; s++) {
    k_gates<<<64, 256, 0, stream>>>(xw, hw, Wih, Whh, bih, bhh, gw);
    k_lstm <<<128, 256, 0, stream>>>(gw, hw, cw);
    k_query<<<16, 256, 0, stream>>>(hw, Wq, bq, qw);
    float* lstep = logits_all + (size_t)s * BATCH * NENC;
    k_attn <<<BATCH * SPLIT, 256, 0, stream>>>(enc, qw, lstep);
    k_topk <<<BATCH, 256, 0, stream>>>(lstep, enc, idx_all + s * BATCH * 3, xw);
  }
}